// InteractionBlock_54382875902596
// MI455X (gfx1250) — compile-verified
//
#include <hip/hip_runtime.h>
#include <hip/hip_bf16.h>

typedef float v2f __attribute__((ext_vector_type(2)));
typedef float v8f __attribute__((ext_vector_type(8)));

__device__ __forceinline__ v8f wmma_f32(v2f a, v2f b, v8f c) {
    // (neg_a, A, neg_b, B, c_mod, C, reuse_a, reuse_b)
    return __builtin_amdgcn_wmma_f32_16x16x4_f32(false, a, false, b, (short)0, c,
                                                 false, false);
}

__device__ __forceinline__ float swish(float v) {
    return v / (1.0f + __expf(-v));
}

#define WAVES 8

// out[n,j] = scale * sum_k A[n,k] * B[k,j];  A:[nrows x 64], B:[64 x 64]
__global__ __launch_bounds__(256) void node_gemm_kernel(
    const float* __restrict__ A, const float* __restrict__ B,
    float* __restrict__ out, int nrows, float scale)
{
    const int wv   = threadIdx.x >> 5;
    const int lane = threadIdx.x & 31;
    const long tile = (long)blockIdx.x * WAVES + wv;
    const int  r0   = (int)(tile * 16);
    if (r0 + 16 > nrows) return;            // wave-uniform

    const bool hi   = lane >= 16;
    const int  l16  = lane & 15;
    const int  m    = l16;
    const int  kofs = hi ? 2 : 0;

    v8f zero = {};
    v8f acc[4];
    for (int ct = 0; ct < 4; ++ct) acc[ct] = zero;

    for (int kt = 0; kt < 16; ++kt) {
        const int k0 = kt * 4 + kofs;
        const float* ap = A + (long)(r0 + m) * 64 + k0;
        v2f a; a.x = ap[0]; a.y = ap[1];
        for (int ct = 0; ct < 4; ++ct) {
            const int n = ct * 16 + l16;
            v2f b; b.x = B[(k0 + 0) * 64 + n]; b.y = B[(k0 + 1) * 64 + n];
            acc[ct] = wmma_f32(a, b, acc[ct]);
        }
    }
    for (int ct = 0; ct < 4; ++ct)
        for (int r = 0; r < 8; ++r)
            out[(long)(r0 + r + (hi ? 8 : 0)) * 64 + ct * 16 + l16] =
                acc[ct][r] * scale;
}

// Fused per-edge pipeline: radial MLP -> w -> contract with Y -> * x[sender]
// -> atomic scatter into agg[receiver].  One wave handles 16 edges.
__global__ __launch_bounds__(256) void edge_kernel(
    const float* __restrict__ x,          // [N,64] up-projected node features
    const float* __restrict__ edge_feat,  // [E,4]
    const float* __restrict__ radial,     // [E,8]
    const int*   __restrict__ senders,    // [E]
    const int*   __restrict__ receivers,  // [E]
    const unsigned char* __restrict__ mask, // [E] bool
    const float* __restrict__ W1,         // [8,64]
    const float* __restrict__ W2,         // [64,64]
    const float* __restrict__ W3,         // [64,256]
    float* __restrict__ agg,              // [N,64]
    int E)
{
    __shared__ float h[WAVES][16][65];    // padded pitch to avoid bank conflicts
    __shared__ float Ysh[WAVES][16][4];   // edge attrs * mask
    __shared__ int   snd[WAVES][16];
    __shared__ int   rcv[WAVES][16];

    const int wv   = threadIdx.x >> 5;
    const int lane = threadIdx.x & 31;
    const long tile = (long)blockIdx.x * WAVES + wv;
    const int  e0   = (int)(tile * 16);
    if (e0 + 16 > E) return;              // wave-uniform

    const bool hi   = lane >= 16;
    const int  l16  = lane & 15;
    const int  m    = l16;                // A-fragment row
    const int  kofs = hi ? 2 : 0;

    // ---- per-edge metadata (same-wave LDS; DS ops are in-order per wave) ----
    if (lane < 16) {
        snd[wv][lane] = senders[e0 + lane];
        rcv[wv][lane] = receivers[e0 + lane];
    }
    for (int i = lane; i < 64; i += 32) {
        const int e = i >> 2, l = i & 3;
        const float mk = mask[e0 + e] ? 1.0f : 0.0f;
        Ysh[wv][e][l] = edge_feat[(long)(e0 + e) * 4 + l] * mk;
    }

    v8f zero = {};
    v8f acc[4];

    // ---- h1 = swish(radial @ W1), K = 8 ----
    for (int ct = 0; ct < 4; ++ct) acc[ct] = zero;
    for (int kt = 0; kt < 2; ++kt) {
        const int k0 = kt * 4 + kofs;
        const float* rp = radial + (long)(e0 + m) * 8 + k0;
        v2f a; a.x = rp[0]; a.y = rp[1];
        for (int ct = 0; ct < 4; ++ct) {
            const int n = ct * 16 + l16;
            v2f b; b.x = W1[(k0 + 0) * 64 + n]; b.y = W1[(k0 + 1) * 64 + n];
            acc[ct] = wmma_f32(a, b, acc[ct]);
        }
    }
    for (int ct = 0; ct < 4; ++ct)
        for (int r = 0; r < 8; ++r)
            h[wv][r + (hi ? 8 : 0)][ct * 16 + l16] = swish(acc[ct][r]);

    // ---- h2 = swish(h1 @ W2), K = 64 ----
    for (int ct = 0; ct < 4; ++ct) acc[ct] = zero;
    for (int kt = 0; kt < 16; ++kt) {
        const int k0 = kt * 4 + kofs;
        v2f a; a.x = h[wv][m][k0]; a.y = h[wv][m][k0 + 1];
        for (int ct = 0; ct < 4; ++ct) {
            const int n = ct * 16 + l16;
            v2f b; b.x = W2[(k0 + 0) * 64 + n]; b.y = W2[(k0 + 1) * 64 + n];
            acc[ct] = wmma_f32(a, b, acc[ct]);
        }
    }
    for (int ct = 0; ct < 4; ++ct)
        for (int r = 0; r < 8; ++r)
            h[wv][r + (hi ? 8 : 0)][ct * 16 + l16] = swish(acc[ct][r]);

    // ---- w = h2 @ W3 (tile-by-tile), fused contraction + scatter ----
    // column c = f*4 + l  ->  l = lane&3,  f = ct*4 + ((lane&15)>>2)
    const int  lq   = lane & 3;
    const int  g    = l16 >> 2;
    const bool lead = (lq == 0);

    for (int ct = 0; ct < 16; ++ct) {
        v8f c = zero;
        const int nn = ct * 16 + l16;
        for (int kt = 0; kt < 16; ++kt) {
            const int k0 = kt * 4 + kofs;
            v2f a; a.x = h[wv][m][k0]; a.y = h[wv][m][k0 + 1];
            v2f b; b.x = W3[(long)(k0 + 0) * 256 + nn];
                   b.y = W3[(long)(k0 + 1) * 256 + nn];
            c = wmma_f32(a, b, c);
        }
        const int f = ct * 4 + g;
        for (int r = 0; r < 8; ++r) {
            const int row = r + (hi ? 8 : 0);
            float v = c[r] * Ysh[wv][row][lq];     // * Y[e,l] (mask folded in)
            v += __shfl_xor(v, 1, 32);             // reduce over l (4 lanes)
            v += __shfl_xor(v, 2, 32);
            if (lead) {
                const float xs = x[(long)snd[wv][row] * 64 + f];
                atomicAdd(&agg[(long)rcv[wv][row] * 64 + f], v * xs);
            }
        }
    }
}

extern "C" void kernel_launch(void* const* d_in, const int* in_sizes, int n_in,
                              void* d_out, int out_size, void* d_ws, size_t ws_size,
                              hipStream_t stream)
{
    const float* node = (const float*)d_in[0];
    const float* ef   = (const float*)d_in[1];
    const float* rad  = (const float*)d_in[2];
    const int*   snd  = (const int*)d_in[3];
    const int*   rcv  = (const int*)d_in[4];
    const unsigned char* mask = (const unsigned char*)d_in[5];
    const float* Wup = (const float*)d_in[6];
    const float* W1  = (const float*)d_in[7];
    const float* W2  = (const float*)d_in[8];
    const float* W3  = (const float*)d_in[9];
    const float* Wdn = (const float*)d_in[10];
    float* out = (float*)d_out;

    const int N = in_sizes[0] / 64;
    const int E = in_sizes[3];

    float* x   = (float*)d_ws;              // [N,64]
    float* agg = x + (size_t)N * 64;        // [N,64]

    hipMemsetAsync(agg, 0, (size_t)N * 64 * sizeof(float), stream);

    const int nodeTiles  = N / 16;                         // 50000/16 = 3125 exact
    const int nodeBlocks = (nodeTiles + WAVES - 1) / WAVES;
    node_gemm_kernel<<<nodeBlocks, 256, 0, stream>>>(node, Wup, x, N, 1.0f);

    const int edgeTiles  = E / 16;                         // 800000/16 exact
    const int edgeBlocks = (edgeTiles + WAVES - 1) / WAVES;
    edge_kernel<<<edgeBlocks, 256, 0, stream>>>(x, ef, rad, snd, rcv, mask,
                                                W1, W2, W3, agg, E);

    node_gemm_kernel<<<nodeBlocks, 256, 0, stream>>>(agg, Wdn, out, N,
                                                     1.0f / 16.0f);
}